// PaiNN_83494164234365
// MI455X (gfx1250) — compile-verified
//
#include <hip/hip_runtime.h>
#include <math.h>

// ---------------------------------------------------------------------------
// PaiNN on MI455X (gfx1250, wave32).
//   * fp32 WMMA (v_wmma_f32_16x16x4_f32) for all dense GEMMs, B panel staged
//     in LDS (bank-conflict-free, ds_load_b64 fragments).
//   * Edge phase folds the K=20 RBF projection in (no 246MB Wr tensor);
//     weights DMA'd into LDS with global_load_async_to_lds_b128 (ASYNCcnt).
//   * Scatter via global_atomic_add_f32. Workload is HBM/scatter bound.
// ---------------------------------------------------------------------------

typedef float v2f __attribute__((ext_vector_type(2)));
typedef float v4f __attribute__((ext_vector_type(4)));
typedef float v8f __attribute__((ext_vector_type(8)));

#define FDIM  128
#define F3    384
#define NRBF_ 20
#define NG_   500
#define KMAX  256
#define KPAD  (KMAX + 4)   // LDS column stride: 4*(k) banks per column pair

__device__ __forceinline__ float silu_f(float x) {
    return x / (1.0f + __expf(-x));
}

// ---------------------------------------------------------------------------
// fp32 WMMA GEMM:  C[M,N] = act(A[M,K] @ B[K,N] + bias[N]),  K <= KMAX.
// 256 threads = 8 waves; block tile = 128 rows x 16 cols; one 16x16 tile
// per wave. B panel (K x 16) staged in LDS column-major so each lane's
// {K=k..k+1 | k+2..k+3} fragment is one ds_load_b64 (pad => no bank conflict).
// A fragment layout (ISA 7.12.2, 32-bit A 16x4): lanes 0-15 K={k,k+1},
// lanes 16-31 K={k+2,k+3}; C/D: VGPR r -> M = r + 8*(lane>=16).
// ---------------------------------------------------------------------------
__global__ __launch_bounds__(256) void painn_gemm_wmma(
    const float* __restrict__ A, const float* __restrict__ B,
    const float* __restrict__ bias, float* __restrict__ C,
    int M, int N, int K, int act)
{
    __shared__ __align__(16) float bs[16 * KPAD];

    const int tid = threadIdx.x;
    const int tn  = blockIdx.x;                 // N/16 tile
    // cooperative stage of B[K x 16] panel, transposed into bs[col][k]
    for (int i = tid; i < (K << 4); i += 256) {
        const int kk = i >> 4;
        const int c  = i & 15;
        bs[c * KPAD + kk] = B[kk * N + (tn << 4) + c];
    }
    __syncthreads();

    const int lane = tid & 31;
    const int wave = tid >> 5;
    const int half = lane >> 4;                 // 0: lanes 0-15, 1: 16-31
    const int l16  = lane & 15;
    const int tm   = blockIdx.y * 8 + wave;     // M/16 tile (uniform per wave)

    if (tm * 16 < M) {
        const int row = (tm << 4) + l16;
        const int col = (tn << 4) + l16;
        const float* Ap = A + row * K + (half << 1);     // +4 per k-step
        const float* Bp = bs + l16 * KPAD + (half << 1); // +4 per k-step

        v8f acc = {};
        #pragma unroll 8
        for (int k = 0; k < K; k += 4) {
            v2f a = *(const v2f*)Ap; Ap += 4;     // global_load_b64
            v2f b = *(const v2f*)Bp; Bp += 4;     // ds_load_b64
            acc = __builtin_amdgcn_wmma_f32_16x16x4_f32(
                      false, a, false, b, (short)0, acc, false, false);
        }

        const float bval = bias ? bias[col] : 0.0f;
        #pragma unroll
        for (int r = 0; r < 8; ++r) {
            const int m = (tm << 4) + r + (half << 3);
            float val = acc[r] + bval;
            if (act) val = silu_f(val);
            C[m * N + col] = val;
        }
    }
}

// ---------------------------------------------------------------------------
// Edge feature: ef[e,n] = sin((n+1)*(3.14/5)*r)/r * (0.5*cos(pi*r/5)+1)
// ---------------------------------------------------------------------------
__global__ void painn_edgefeat(const float* __restrict__ r_ij,
                               float* __restrict__ ef, int n_edges)
{
    int idx = blockIdx.x * blockDim.x + threadIdx.x;
    if (idx >= n_edges * NRBF_) return;
    int e = idx / NRBF_, n = idx % NRBF_;
    float r    = r_ij[e];
    float rbf  = __sinf((float)(n + 1) * (3.14f / 5.0f) * r) / r;
    float fcut = 0.5f * __cosf(3.14159265358979f * r / 5.0f) + 1.0f;
    ef[idx] = rbf * fcut;
}

// ---------------------------------------------------------------------------
// Edge message kernel. Fuses Wr = ef @ w_w + w_b (K=20, weights in LDS via
// async-DMA), conv = h[src]*Wr, and the ds/dv scatter-adds.
// v_old is a pre-update snapshot of v (reference reads old v).
// ---------------------------------------------------------------------------
__global__ __launch_bounds__(128) void painn_edge(
    const float* __restrict__ ef, const float* __restrict__ r_dir,
    const int* __restrict__ edges, const float* __restrict__ h,
    const float* __restrict__ ww, const float* __restrict__ wb,
    const float* __restrict__ v_old,
    float* __restrict__ s, float* __restrict__ v, int n_edges)
{
    __shared__ __align__(16) float sww[NRBF_ * F3];   // 30720 B
    __shared__ __align__(16) float swb[F3];
    const int t = threadIdx.x;          // 0..127

    // Async DMA global -> LDS (gfx1250 ASYNCcnt path), 16B per op.
    for (int i = 4 * t; i < NRBF_ * F3; i += 4 * 128) {
        unsigned lds = (unsigned)(unsigned long long)&sww[i];
        asm volatile("global_load_async_to_lds_b128 %0, %1, off"
                     :: "v"(lds), "v"(ww + i) : "memory");
    }
    if (t < F3 / 4) {
        unsigned lds = (unsigned)(unsigned long long)&swb[4 * t];
        asm volatile("global_load_async_to_lds_b128 %0, %1, off"
                     :: "v"(lds), "v"(wb + 4 * t) : "memory");
    }
    asm volatile("s_wait_asynccnt 0" ::: "memory");
    __syncthreads();

    for (int e = blockIdx.x; e < n_edges; e += gridDim.x) {
        const int src = edges[2 * e + 0];
        const int dst = edges[2 * e + 1];

        // 20 RBF coefficients: edge stride is 80B (16-aligned) -> 5x b128
        float efr[NRBF_];
        const v4f* efp = (const v4f*)(ef + e * NRBF_);
        #pragma unroll
        for (int q = 0; q < NRBF_ / 4; ++q) {
            v4f x = efp[q];
            efr[4 * q + 0] = x.x; efr[4 * q + 1] = x.y;
            efr[4 * q + 2] = x.z; efr[4 * q + 3] = x.w;
        }

        float a0 = swb[t], a1 = swb[FDIM + t], a2 = swb[2 * FDIM + t];
        #pragma unroll
        for (int k = 0; k < NRBF_; ++k) {
            const float ek = efr[k];
            a0 += ek * sww[k * F3 + t];
            a1 += ek * sww[k * F3 + FDIM + t];
            a2 += ek * sww[k * F3 + 2 * FDIM + t];
        }
        const int hb = src * F3;
        const float dvg  = h[hb + t]            * a0;  // conv[:,0:128]
        const float dsv  = h[hb + FDIM + t]     * a1;  // conv[:,128:256]
        const float drep = h[hb + 2 * FDIM + t] * a2;  // conv[:,256:384]

        atomicAdd(&s[dst * FDIM + t], dsv);
        #pragma unroll
        for (int d = 0; d < 3; ++d) {
            float dv = v_old[(src * 3 + d) * FDIM + t] * dvg
                     + r_dir[e * 3 + d] * drep;
            atomicAdd(&v[(dst * 3 + d) * FDIM + t], dv);
        }
    }
}

// vn = sqrt(sum_d Vv^2 + 1e-12); cat = [s, vn]
__global__ void painn_vncat(const float* __restrict__ Vv,
                            const float* __restrict__ s,
                            float* __restrict__ cat, int n_atoms)
{
    int idx = blockIdx.x * blockDim.x + threadIdx.x;
    if (idx >= n_atoms * FDIM) return;
    int i = idx >> 7, f = idx & 127;
    float x0 = Vv[(i * 3 + 0) * FDIM + f];
    float x1 = Vv[(i * 3 + 1) * FDIM + f];
    float x2 = Vv[(i * 3 + 2) * FDIM + f];
    float vn = sqrtf(x0 * x0 + x1 * x1 + x2 * x2 + 1e-12f);
    cat[i * 2 * FDIM + f]        = s[idx];
    cat[i * 2 * FDIM + FDIM + f] = vn;
}

// s += ass + asv * <Uv,Vv>_d ;  v += avv * Uv
__global__ void painn_update(const float* __restrict__ a,
                             const float* __restrict__ Uv,
                             const float* __restrict__ Vv,
                             float* __restrict__ s, float* __restrict__ v,
                             int n_atoms)
{
    int idx = blockIdx.x * blockDim.x + threadIdx.x;
    if (idx >= n_atoms * FDIM) return;
    int i = idx >> 7, f = idx & 127;
    const int ab = i * F3;
    float avv = a[ab + f];
    float asv = a[ab + FDIM + f];
    float ass = a[ab + 2 * FDIM + f];
    float dot = 0.0f;
    #pragma unroll
    for (int d = 0; d < 3; ++d)
        dot += Uv[(i * 3 + d) * FDIM + f] * Vv[(i * 3 + d) * FDIM + f];
    s[idx] += ass + asv * dot;
    #pragma unroll
    for (int d = 0; d < 3; ++d)
        v[(i * 3 + d) * FDIM + f] += avv * Uv[(i * 3 + d) * FDIM + f];
}

__global__ void painn_init_s(const float* __restrict__ emb,
                             const int* __restrict__ z,
                             float* __restrict__ s, int n_atoms)
{
    int idx = blockIdx.x * blockDim.x + threadIdx.x;
    if (idx >= n_atoms * FDIM) return;
    int i = idx >> 7, f = idx & 127;
    s[idx] = emb[z[i] * FDIM + f];
}

__global__ void painn_zero(float* __restrict__ p, int n)
{
    int i = blockIdx.x * blockDim.x + threadIdx.x;
    if (i < n) p[i] = 0.0f;
}

// atom_e = e1 @ out_w2 + b2, segment-summed into out[graph_idx].
// One wave32 per atom; shfl_xor tree reduce; lane 0 does the atomic.
__global__ __launch_bounds__(128) void painn_reduce(
    const float* __restrict__ e1, const float* __restrict__ w2,
    const float* __restrict__ b2, const int* __restrict__ gidx,
    float* __restrict__ out, int n_atoms)
{
    const int wave = threadIdx.x >> 5;
    const int lane = threadIdx.x & 31;
    const int atom = blockIdx.x * 4 + wave;
    if (atom >= n_atoms) return;
    float sum = 0.0f;
    #pragma unroll
    for (int j = 0; j < FDIM / 32; ++j)
        sum += e1[atom * FDIM + lane + 32 * j] * w2[lane + 32 * j];
    #pragma unroll
    for (int off = 16; off > 0; off >>= 1)
        sum += __shfl_xor(sum, off, 32);
    if (lane == 0) atomicAdd(&out[gidx[atom]], sum + b2[0]);
}

// ---------------------------------------------------------------------------
extern "C" void kernel_launch(void* const* d_in, const int* in_sizes, int n_in,
                              void* d_out, int out_size, void* d_ws, size_t ws_size,
                              hipStream_t stream)
{
    const float* r_ij   = (const float*)d_in[0];
    const float* r_dir  = (const float*)d_in[1];
    const int*   z      = (const int*)  d_in[2];
    const int*   edges  = (const int*)  d_in[3];
    const int*   gidx   = (const int*)  d_in[4];
    const float* emb    = (const float*)d_in[5];
    const float* phi_w1 = (const float*)d_in[6];
    const float* phi_b1 = (const float*)d_in[7];
    const float* phi_w2 = (const float*)d_in[8];
    const float* phi_b2 = (const float*)d_in[9];
    const float* w_w    = (const float*)d_in[10];
    const float* w_b    = (const float*)d_in[11];
    const float* U      = (const float*)d_in[12];
    const float* V      = (const float*)d_in[13];
    const float* up_w1  = (const float*)d_in[14];
    const float* up_b1  = (const float*)d_in[15];
    const float* up_w2  = (const float*)d_in[16];
    const float* up_b2  = (const float*)d_in[17];
    const float* out_w1 = (const float*)d_in[18];
    const float* out_b1 = (const float*)d_in[19];
    const float* out_w2 = (const float*)d_in[20];
    const float* out_b2 = (const float*)d_in[21];
    float* out = (float*)d_out;

    const int NATOM = in_sizes[2];       // 10000
    const int NEDGE = in_sizes[0];       // 160000
    const int NB    = 3;

    // workspace carve-up (floats)
    float* ws = (float*)d_ws;
    size_t o = 0;
    float* s    = ws + o; o += (size_t)NATOM * FDIM;
    float* v    = ws + o; o += (size_t)NATOM * F3;
    float* vold = ws + o; o += (size_t)NATOM * F3;
    float* ef   = ws + o; o += (size_t)NEDGE * NRBF_;
    float* t1   = ws + o; o += (size_t)NATOM * FDIM;      // h1 / a1 / e1
    float* t3   = ws + o; o += (size_t)NATOM * F3;        // h  / a
    float* Uv   = ws + o; o += (size_t)NATOM * F3;
    float* Vv   = ws + o; o += (size_t)NATOM * F3;
    float* cat  = ws + o; o += (size_t)NATOM * 2 * FDIM;
    (void)ws_size; (void)n_in; (void)out_size;

    const dim3 blk256(256), blk128(128);
    const int nf = NATOM * FDIM;
    const int nv = NATOM * F3;
    auto g1 = [](int n) { return dim3((n + 255) / 256); };

    // GEMM grids: grid.x = N/16 tiles, grid.y covers M/16 tiles, 8 waves/blk
    const dim3 gA(FDIM / 16, (NATOM / 16 + 7) / 8);       // M=10000, N=128
    const dim3 gA3(F3 / 16,  (NATOM / 16 + 7) / 8);       // M=10000, N=384
    const dim3 gV(FDIM / 16, (3 * NATOM / 16 + 7) / 8);   // M=30000, N=128

    painn_zero<<<g1(NG_), blk256, 0, stream>>>(out, NG_);
    painn_zero<<<g1(nv),  blk256, 0, stream>>>(v, nv);
    painn_init_s<<<g1(nf), blk256, 0, stream>>>(emb, z, s, NATOM);
    painn_edgefeat<<<g1(NEDGE * NRBF_), blk256, 0, stream>>>(r_ij, ef, NEDGE);

    for (int b = 0; b < NB; ++b) {
        // h = silu(s @ phi_w1 + b1) @ phi_w2 + b2
        painn_gemm_wmma<<<gA, blk256, 0, stream>>>(
            s, phi_w1 + (size_t)b * FDIM * FDIM, phi_b1 + (size_t)b * FDIM,
            t1, NATOM, FDIM, FDIM, 1);
        painn_gemm_wmma<<<gA3, blk256, 0, stream>>>(
            t1, phi_w2 + (size_t)b * FDIM * F3, phi_b2 + (size_t)b * F3,
            t3, NATOM, F3, FDIM, 0);

        // snapshot v (edge phase must read pre-update v), then message pass
        hipMemcpyAsync(vold, v, (size_t)nv * sizeof(float),
                       hipMemcpyDeviceToDevice, stream);
        painn_edge<<<dim3(2000), blk128, 0, stream>>>(
            ef, r_dir, edges, t3,
            w_w + (size_t)b * NRBF_ * F3, w_b + (size_t)b * F3,
            vold, s, v, NEDGE);

        // Uv, Vv  (30000x128 @ 128x128)
        painn_gemm_wmma<<<gV, blk256, 0, stream>>>(
            v, U + (size_t)b * FDIM * FDIM, nullptr, Uv, 3 * NATOM, FDIM, FDIM, 0);
        painn_gemm_wmma<<<gV, blk256, 0, stream>>>(
            v, V + (size_t)b * FDIM * FDIM, nullptr, Vv, 3 * NATOM, FDIM, FDIM, 0);

        painn_vncat<<<g1(nf), blk256, 0, stream>>>(Vv, s, cat, NATOM);

        // a = silu([s,vn] @ up_w1 + b1) @ up_w2 + b2
        painn_gemm_wmma<<<gA, blk256, 0, stream>>>(
            cat, up_w1 + (size_t)b * 2 * FDIM * FDIM, up_b1 + (size_t)b * FDIM,
            t1, NATOM, FDIM, 2 * FDIM, 1);
        painn_gemm_wmma<<<gA3, blk256, 0, stream>>>(
            t1, up_w2 + (size_t)b * FDIM * F3, up_b2 + (size_t)b * F3,
            t3, NATOM, F3, FDIM, 0);

        painn_update<<<g1(nf), blk256, 0, stream>>>(t3, Uv, Vv, s, v, NATOM);
    }

    // atom energies + per-graph segment sum
    painn_gemm_wmma<<<gA, blk256, 0, stream>>>(
        s, out_w1, out_b1, t1, NATOM, FDIM, FDIM, 1);
    painn_reduce<<<dim3((NATOM + 3) / 4), blk128, 0, stream>>>(
        t1, out_w2, out_b2, gidx, out, NATOM);
}